// ClusterisationLoss_21930103013687
// MI455X (gfx1250) — compile-verified
//
#include <hip/hip_runtime.h>
#include <hip/hip_bf16.h>

typedef __attribute__((ext_vector_type(2))) float v2f;
typedef __attribute__((ext_vector_type(8))) float v8f;

#define NROWS   8192
#define IN_DIM  256
#define NCLS    64
#define MARGIN_F 0.5f
#define EPS_F    1e-6f
#define LDA     68            // LDS row stride (floats): 68 % 64 == 4 -> no bank conflicts

static __device__ __forceinline__ v8f wmma_f32_k4(v2f a, v2f b, v8f c) {
  // D = A(16x4,f32) * B(4x16,f32) + C(16x16,f32)   -> v_wmma_f32_16x16x4_f32
  return __builtin_amdgcn_wmma_f32_16x16x4_f32(false, a, false, b, (short)0, c,
                                               false, false);
}

// ---------------------------------------------------------------------------
// Stage 1: e = E @ W^T + b       [8192,256] x [64,256]^T -> [8192,64]
// One wave per 16x16 output tile; fp32 WMMA, K loop of 4.
// ---------------------------------------------------------------------------
__global__ __launch_bounds__(256) void fc_kernel(const float* __restrict__ E,
                                                 const float* __restrict__ W,
                                                 const float* __restrict__ bias,
                                                 float* __restrict__ e)
{
  const int wave    = (blockIdx.x * 256 + (int)threadIdx.x) >> 5;
  const int lane    = threadIdx.x & 31;
  const int rowTile = wave >> 2;            // 512 row tiles
  const int colTile = wave & 3;             // 4 col tiles
  const int li      = lane & 15;
  const int kh      = (lane >> 4) << 1;     // 0 or 2

  const float* Arow = E + (size_t)(rowTile * 16 + li) * IN_DIM;
  const float* Brow = W + (size_t)(colTile * 16 + li) * IN_DIM;

  v8f acc = {};
  for (int k0 = 0; k0 < IN_DIM; k0 += 4) {
    v2f a, b;
    a.x = Arow[k0 + kh]; a.y = Arow[k0 + kh + 1];
    b.x = Brow[k0 + kh]; b.y = Brow[k0 + kh + 1];
    acc = wmma_f32_k4(a, b, acc);
  }

  const int   n     = colTile * 16 + li;
  const float bn    = bias[n];
  const int   ibase = rowTile * 16 + ((lane >> 4) << 3);
#pragma unroll
  for (int r = 0; r < 8; ++r)
    e[(size_t)(ibase + r) * NCLS + n] = acc[r] + bn;
}

// ---------------------------------------------------------------------------
// Stage 2: per-row argmax -> labels (first-max semantics, strict >)
// ---------------------------------------------------------------------------
__global__ void argmax_kernel(const float* __restrict__ e, int* __restrict__ lbl)
{
  const int i = blockIdx.x * blockDim.x + threadIdx.x;
  const float* row = e + (size_t)i * NCLS;
  float best = row[0]; int bi = 0;
  for (int c = 1; c < NCLS; ++c) {
    float v = row[c];
    if (v > best) { best = v; bi = c; }
  }
  lbl[i] = bi;
}

// ---------------------------------------------------------------------------
// Stage 3: deterministic per-class sums + counts (block c reduces class c)
// ---------------------------------------------------------------------------
__global__ __launch_bounds__(256) void clustersum_kernel(const float* __restrict__ e,
                                                         const int*   __restrict__ lbl,
                                                         float* __restrict__ sums,
                                                         float* __restrict__ counts)
{
  const int c = blockIdx.x;
  const int m = threadIdx.x & 63;
  const int g = threadIdx.x >> 6;           // 0..3 row-strides
  float acc = 0.f, cnt = 0.f;
  for (int i = g; i < NROWS; i += 4) {
    const bool in = (lbl[i] == c);
    if (in) acc += e[(size_t)i * NCLS + m];
    if (m == 0 && in) cnt += 1.f;
  }
  __shared__ float sm[4][64];
  __shared__ float sc[4];
  sm[g][m] = acc;
  if (m == 0) sc[g] = cnt;
  __syncthreads();
  if (g == 0) {
    sums[c * NCLS + m] = sm[0][m] + sm[1][m] + sm[2][m] + sm[3][m];
    if (m == 0) counts[c] = sc[0] + sc[1] + sc[2] + sc[3];
  }
}

// ---------------------------------------------------------------------------
// Stage 4: means (in place) + per-class weight 1/(w3*C)
// ---------------------------------------------------------------------------
__global__ void finalize_kernel(float* __restrict__ sums,
                                const float* __restrict__ counts,
                                float* __restrict__ invw3c)
{
  const int idx = blockIdx.x * blockDim.x + threadIdx.x;   // 0..4095
  const int c   = idx >> 6;
  const float w = counts[c];
  const float wd = (w == 0.f) ? 1.f : w;
  sums[idx] = sums[idx] / wd;
  if (idx < NCLS) {
    const float w2 = counts[idx] - 1.f;
    const float w3 = (w2 <= 0.f) ? 1.f : w2;
    invw3c[idx] = 1.f / (w3 * (float)NCLS);
  }
}

// ---------------------------------------------------------------------------
// Stage 5: negative loss = relu(margin - min_{active pairs} ||mu_a-mu_b+eps||^2)^2
// ---------------------------------------------------------------------------
__global__ __launch_bounds__(256) void negative_kernel(const float* __restrict__ means,
                                                       const float* __restrict__ counts,
                                                       float* __restrict__ out)
{
  float dmin = 3.0e38f;
  for (int p = threadIdx.x; p < NCLS * NCLS; p += 256) {
    const int a = p >> 6, b = p & 63;
    if (a == b) continue;
    if (counts[a] == 0.f || counts[b] == 0.f) continue;
    float d2 = 0.f;
    for (int m = 0; m < NCLS; ++m) {
      const float dm = means[a * NCLS + m] - means[b * NCLS + m] + EPS_F;
      d2 += dm * dm;
    }
    dmin = fminf(dmin, d2);
  }
  __shared__ float sm[256];
  sm[threadIdx.x] = dmin;
  __syncthreads();
  for (int off = 128; off > 0; off >>= 1) {
    if ((int)threadIdx.x < off) sm[threadIdx.x] = fminf(sm[threadIdx.x], sm[threadIdx.x + off]);
    __syncthreads();
  }
  if (threadIdx.x == 0) {
    int nact = 0;
    for (int c = 0; c < NCLS; ++c) nact += (counts[c] != 0.f) ? 1 : 0;
    float neg = 0.f;
    if (nact > 1) {
      const float r = fmaxf(MARGIN_F - sm[0], 0.f);
      neg = r * r;
    }
    out[1] = neg;
  }
}

// ---------------------------------------------------------------------------
// Stage 6: e2 = e - means[lbl], plus per-row sq = ||e2||^2 and s = sum(e2)
// ---------------------------------------------------------------------------
__global__ void center_kernel(const float* __restrict__ e,
                              const int*   __restrict__ lbl,
                              const float* __restrict__ means,
                              float* __restrict__ e2,
                              float* __restrict__ sq,
                              float* __restrict__ s)
{
  const int i = blockIdx.x * blockDim.x + threadIdx.x;
  const float* row = e + (size_t)i * NCLS;
  const float* mu  = means + (size_t)lbl[i] * NCLS;
  float ssq = 0.f, ssum = 0.f;
  for (int m = 0; m < NCLS; ++m) {
    const float v = row[m] - mu[m];
    e2[(size_t)i * NCLS + m] = v;
    ssq += v * v; ssum += v;
  }
  sq[i] = ssq; s[i] = ssum;
}

// ---------------------------------------------------------------------------
// Stage 7: pairwise masked distance sum, LDS-tiled 128x128 per block.
//  - coalesced float4 staging of two 128x64 slabs into padded LDS (stride 68)
//  - A fragment hoisted to 32 VGPRs; inner loop = ds_load_b64(B) + fp32 WMMA
//  - 128 v_wmma_f32_16x16x4_f32 per wave; epilogue masked weighted distance
//  - deterministic wave/block reduction into partials[]
// ---------------------------------------------------------------------------
__global__ __launch_bounds__(256) void pairwise_kernel(const float* __restrict__ e2,
                                                       const float* __restrict__ sq,
                                                       const float* __restrict__ s,
                                                       const int*   __restrict__ lbl,
                                                       const float* __restrict__ invw3c,
                                                       float* __restrict__ partials)
{
  extern __shared__ char smem[];
  float* As     = (float*)smem;            // 128 * LDA
  float* Bs     = As + 128 * LDA;          // 128 * LDA
  float* sqi_s  = Bs + 128 * LDA;          // 128
  float* si_s   = sqi_s + 128;             // 128
  float* sqj_s  = si_s + 128;              // 128
  float* sj_s   = sqj_s + 128;             // 128
  float* iw_s   = sj_s + 128;              // 64
  int*   lbli_s = (int*)(iw_s + 64);       // 128
  int*   lblj_s = lbli_s + 128;            // 128

  const int tid   = threadIdx.x;
  const int iBase = (int)(blockIdx.x >> 6) * 128;
  const int jBase = (int)(blockIdx.x & 63) * 128;

  // Stage the two 128x64 fp32 slabs with coalesced float4 loads (16 f4 / row).
  for (int idx = tid; idx < 128 * 16; idx += 256) {
    const int row = idx >> 4, q = idx & 15;
    ((float4*)(As + row * LDA))[q] =
        ((const float4*)(e2 + (size_t)(iBase + row) * NCLS))[q];
    ((float4*)(Bs + row * LDA))[q] =
        ((const float4*)(e2 + (size_t)(jBase + row) * NCLS))[q];
  }
  if (tid < 128) {
    sqi_s[tid]  = sq[iBase + tid];
    si_s[tid]   = s[iBase + tid];
    lbli_s[tid] = lbl[iBase + tid];
  } else {
    const int t = tid - 128;
    sqj_s[t]  = sq[jBase + t];
    sj_s[t]   = s[jBase + t];
    lblj_s[t] = lbl[jBase + t];
  }
  if (tid < 64) iw_s[tid] = invw3c[tid];
  __syncthreads();

  const int w    = tid >> 5;               // wave id: i-subtile
  const int lane = tid & 31;
  const int li   = lane & 15;
  const int kh   = (lane >> 4) << 1;

  // Hoist the A fragment (loop-invariant over j-subtiles): 16 x v2f = 32 VGPRs.
  v2f aReg[16];
  {
    const float* ap = As + (w * 16 + li) * LDA + kh;
#pragma unroll
    for (int kk = 0; kk < 16; ++kk) {
      aReg[kk].x = ap[kk * 4];
      aReg[kk].y = ap[kk * 4 + 1];
    }
  }

  const float meps2 = (float)NCLS * EPS_F * EPS_F;
  const int   ibl   = w * 16 + ((lane >> 4) << 3);
  float part = 0.f;

  for (int jsub = 0; jsub < 8; ++jsub) {
    const float* bp = Bs + (jsub * 16 + li) * LDA + kh;
    v8f acc = {};
#pragma unroll
    for (int kk = 0; kk < 16; ++kk) {
      v2f b;
      b.x = bp[kk * 4];
      b.y = bp[kk * 4 + 1];
      acc = wmma_f32_k4(aReg[kk], b, acc);
    }
    // Epilogue: lane L vgpr r -> local i = ibl + r, local j = jsub*16 + L%16
    const int   jl  = jsub * 16 + li;
    const float sqj = sqj_s[jl];
    const float sj  = sj_s[jl];
    const int   ljv = lblj_s[jl];
#pragma unroll
    for (int r = 0; r < 8; ++r) {
      const int il = ibl + r;
      const float d2 = sqi_s[il] + sqj - 2.f * acc[r]
                     + 2.f * EPS_F * (si_s[il] - sj) + meps2;
      const float d  = __builtin_sqrtf(fmaxf(d2, 1e-12f));
      const int lli  = lbli_s[il];
      part += (lli == ljv) ? d * iw_s[lli] : 0.f;
    }
  }

  // wave32 reduction (fixed order -> deterministic)
  for (int off = 16; off > 0; off >>= 1)
    part += __shfl_down(part, off, 32);

  __shared__ float wsum[8];
  if (lane == 0) wsum[w] = part;
  __syncthreads();
  if (tid == 0) {
    float t = 0.f;
#pragma unroll
    for (int x = 0; x < 8; ++x) t += wsum[x];
    partials[blockIdx.x] = t;
  }
}

// ---------------------------------------------------------------------------
// Stage 8: deterministic final reduction -> d_out[0]
// ---------------------------------------------------------------------------
__global__ __launch_bounds__(256) void reduce_kernel(const float* __restrict__ partials,
                                                     int n, float* __restrict__ out)
{
  float acc = 0.f;
  for (int i = threadIdx.x; i < n; i += 256) acc += partials[i];
  __shared__ float sm[256];
  sm[threadIdx.x] = acc;
  __syncthreads();
  for (int off = 128; off > 0; off >>= 1) {
    if ((int)threadIdx.x < off) sm[threadIdx.x] += sm[threadIdx.x + off];
    __syncthreads();
  }
  if (threadIdx.x == 0) out[0] = sm[0];
}

// ---------------------------------------------------------------------------
extern "C" void kernel_launch(void* const* d_in, const int* in_sizes, int n_in,
                              void* d_out, int out_size, void* d_ws, size_t ws_size,
                              hipStream_t stream)
{
  (void)in_sizes; (void)n_in; (void)out_size; (void)ws_size;
  const float* E    = (const float*)d_in[0];
  const float* W    = (const float*)d_in[1];
  const float* bias = (const float*)d_in[2];
  float* out = (float*)d_out;

  char*  ws  = (char*)d_ws;
  size_t off = 0;
  auto carve = [&](size_t bytes) -> void* {
    void* p = ws + off;
    off += (bytes + 255) & ~(size_t)255;
    return p;
  };

  float* e      = (float*)carve((size_t)NROWS * NCLS * sizeof(float)); // 2 MB
  float* e2b    = (float*)carve((size_t)NROWS * NCLS * sizeof(float)); // 2 MB
  int*   lbl    = (int*)  carve((size_t)NROWS * sizeof(int));
  float* sq     = (float*)carve((size_t)NROWS * sizeof(float));
  float* s      = (float*)carve((size_t)NROWS * sizeof(float));
  float* sums   = (float*)carve((size_t)NCLS * NCLS * sizeof(float));  // -> means
  float* counts = (float*)carve((size_t)NCLS * sizeof(float));
  float* invw3c = (float*)carve((size_t)NCLS * sizeof(float));
  const int nPairBlocks = (NROWS / 128) * (NROWS / 128);               // 4096
  float* partials = (float*)carve((size_t)nPairBlocks * sizeof(float));

  // dynamic LDS for pairwise: 2 slabs (128*68 f32) + 4*128 f32 + 64 f32 + 2*128 i32
  const size_t smemBytes = (size_t)(2 * 128 * LDA + 4 * 128 + 64) * sizeof(float)
                         + (size_t)(2 * 128) * sizeof(int);            // 72,960 B

  fc_kernel        <<<(NROWS/16)*(NCLS/16)/8, 256, 0, stream>>>(E, W, bias, e);
  argmax_kernel    <<<NROWS/256, 256, 0, stream>>>(e, lbl);
  clustersum_kernel<<<NCLS, 256, 0, stream>>>(e, lbl, sums, counts);
  finalize_kernel  <<<(NCLS*NCLS)/256, 256, 0, stream>>>(sums, counts, invw3c);
  negative_kernel  <<<1, 256, 0, stream>>>(sums, counts, out);
  center_kernel    <<<NROWS/256, 256, 0, stream>>>(e, lbl, sums, e2b, sq, s);
  pairwise_kernel  <<<nPairBlocks, 256, smemBytes, stream>>>(e2b, sq, s, lbl, invw3c, partials);
  reduce_kernel    <<<1, 256, 0, stream>>>(partials, nPairBlocks, out);
}